// SimpleGCN_16149077033111
// MI455X (gfx1250) — compile-verified
//
#include <hip/hip_runtime.h>

#define NNODES 10000
#define DM 128

typedef float v2f __attribute__((ext_vector_type(2)));
typedef float v8f __attribute__((ext_vector_type(8)));

// ---------------- degree / normalization ----------------

__global__ __launch_bounds__(256) void k_deg_init(unsigned int* deg, int n) {
    int i = blockIdx.x * blockDim.x + threadIdx.x;
    if (i < n) deg[i] = 1u;  // self-loop contributes 1
}

__global__ __launch_bounds__(256) void k_deg_accum(const int* __restrict__ col,
                                                   unsigned int* deg, int E) {
    int i = blockIdx.x * blockDim.x + threadIdx.x;
    if (i < E) atomicAdd(&deg[col[i]], 1u);
}

__global__ __launch_bounds__(256) void k_dis(const unsigned int* __restrict__ deg,
                                             float* __restrict__ dis, int n) {
    int i = blockIdx.x * blockDim.x + threadIdx.x;
    if (i < n) dis[i] = rsqrtf((float)deg[i]);  // deg >= 1 always
}

// ---------------- dense GEMM via fp32 WMMA ----------------
// H[M,128] = A[M,128] @ W[128,128]; M multiple of 16.
// Block = 256 threads = 8 waves; wave w computes the 16x16 tile at
// (m0 = 16*blockIdx.x, n0 = 16*w). K-loop: 32 x V_WMMA_F32_16X16X4_F32.
__global__ __launch_bounds__(256) void k_gemm_wmma(const float* __restrict__ A,
                                                   const float* __restrict__ W,
                                                   float* __restrict__ H, int M) {
    const int wave = threadIdx.x >> 5;
    const int lane = threadIdx.x & 31;
    const int m0 = blockIdx.x * 16;
    const int n0 = wave * 16;
    if (m0 >= M) return;

    const int mr = lane & 15;        // M (for A) / N (for B,C)
    const int kh = (lane >> 4) * 2;  // K pair selected by lane half

    const float* Arow = A + (size_t)(m0 + mr) * DM;  // A: row m0+mr
    v8f c = {};

#pragma unroll
    for (int k0 = 0; k0 < DM; k0 += 4) {
        v2f a, b;
        // A 16x4 layout: vgpr0 -> K=kh, vgpr1 -> K=kh+1 (kh = 0 or 2 by lane half)
        a.x = Arow[k0 + kh];
        a.y = Arow[k0 + kh + 1];
        // B 4x16 layout (mirror of A): vgpr0 -> K=kh row, vgpr1 -> K=kh+1 row
        b.x = W[(size_t)(k0 + kh) * DM + n0 + mr];
        b.y = W[(size_t)(k0 + kh + 1) * DM + n0 + mr];
        c = __builtin_amdgcn_wmma_f32_16x16x4_f32(false, a, false, b,
                                                  (short)0, c, false, false);
    }

    // C/D layout: vgpr v -> row m0 + v + 8*(lane>>4), col n0 + (lane&15)
    const int rbase = m0 + (lane >> 4) * 8;
    const int cidx = n0 + mr;
#pragma unroll
    for (int v = 0; v < 8; ++v)
        H[(size_t)(rbase + v) * DM + cidx] = c[v];
}

// ---------------- accumulator init: self-loop msg + bias ----------------
__global__ __launch_bounds__(256) void k_init_acc(const float* __restrict__ h,
                                                  const float* __restrict__ dis,
                                                  const float* __restrict__ bias,
                                                  float* __restrict__ acc, int n) {
    int idx = blockIdx.x * blockDim.x + threadIdx.x;
    if (idx < n * DM) {
        int node = idx >> 7;
        int d = idx & (DM - 1);
        float s = dis[node];
        acc[idx] = h[idx] * s * s + bias[d];
    }
}

// ---------------- edge scatter: one wave per edge ----------------
// lane j handles features [4j, 4j+4): float4 gather + 4 native f32 atomics.
__global__ __launch_bounds__(256) void k_scatter(const int* __restrict__ row,
                                                 const int* __restrict__ col,
                                                 const float* __restrict__ dis,
                                                 const float* __restrict__ h,
                                                 float* __restrict__ acc, int E) {
    int w = (int)((blockIdx.x * blockDim.x + threadIdx.x) >> 5);
    int lane = threadIdx.x & 31;
    if (w >= E) return;
    int r = row[w];
    int c = col[w];
    float nrm = dis[r] * dis[c];
    const float4* hv = (const float4*)(h + (size_t)r * DM);
    float4 v = hv[lane];
    float* dst = acc + (size_t)c * DM + lane * 4;
    unsafeAtomicAdd(dst + 0, v.x * nrm);
    unsafeAtomicAdd(dst + 1, v.y * nrm);
    unsafeAtomicAdd(dst + 2, v.z * nrm);
    unsafeAtomicAdd(dst + 3, v.w * nrm);
}

// ---------------- finalize (optional ReLU) ----------------
__global__ __launch_bounds__(256) void k_final(const float* __restrict__ acc,
                                               float* __restrict__ out, int n,
                                               int do_relu) {
    int idx = blockIdx.x * blockDim.x + threadIdx.x;
    if (idx < n * DM) {
        float v = acc[idx];
        out[idx] = (do_relu && v < 0.0f) ? 0.0f : v;
    }
}

extern "C" void kernel_launch(void* const* d_in, const int* in_sizes, int n_in,
                              void* d_out, int out_size, void* d_ws, size_t ws_size,
                              hipStream_t stream) {
    const float* x  = (const float*)d_in[0];
    const int*   ei = (const int*)d_in[1];
    const float* W1 = (const float*)d_in[2];
    const float* b1 = (const float*)d_in[3];
    const float* W2 = (const float*)d_in[4];
    const float* b2 = (const float*)d_in[5];

    const int n = NNODES;
    const int E = in_sizes[1] / 2;
    const int* row = ei;
    const int* col = ei + E;

    float* out_x2 = (float*)d_out;                         // first tuple element
    float* out_x1 = (float*)d_out + (size_t)n * DM;        // second tuple element

    // workspace layout (floats): [deg:16384][dis:16384][h:n*128][acc:n*128]
    float* f = (float*)d_ws;
    unsigned int* deg = (unsigned int*)f;
    float* dis = f + 16384;
    float* h   = f + 2 * 16384;
    float* acc = h + (size_t)n * DM;

    const int B = 256;
    const int gN   = (n + B - 1) / B;            // node-wise
    const int gE   = (E + B - 1) / B;            // edge-wise (thread per edge)
    const int gND  = (n * DM + B - 1) / B;       // node*feature
    const int gEW  = (E + 7) / 8;                // wave per edge, 8 waves/block
    const int gM   = (n + 15) / 16;              // 16-row GEMM tiles

    // normalization
    k_deg_init<<<gN, B, 0, stream>>>(deg, n);
    k_deg_accum<<<gE, B, 0, stream>>>(col, deg, E);
    k_dis<<<gN, B, 0, stream>>>(deg, dis, n);

    // layer 1: x1 = relu(scatter(norm * (xW1)[row] -> col) + b1)
    k_gemm_wmma<<<gM, B, 0, stream>>>(x, W1, h, n);
    k_init_acc<<<gND, B, 0, stream>>>(h, dis, b1, acc, n);
    k_scatter<<<gEW, B, 0, stream>>>(row, col, dis, h, acc, E);
    k_final<<<gND, B, 0, stream>>>(acc, out_x1, n, 1);

    // layer 2: x2 = scatter(norm * (x1 W2)[row] -> col) + b2
    k_gemm_wmma<<<gM, B, 0, stream>>>(out_x1, W2, h, n);
    k_init_acc<<<gND, B, 0, stream>>>(h, dis, b2, acc, n);
    k_scatter<<<gEW, B, 0, stream>>>(row, col, dis, h, acc, E);
    k_final<<<gND, B, 0, stream>>>(acc, out_x2, n, 0);
}